// NGCF_10118942950095
// MI455X (gfx1250) — compile-verified
//
#include <hip/hip_runtime.h>
#include <hip/hip_bf16.h>

typedef float v2f __attribute__((ext_vector_type(2)));
typedef float v8f __attribute__((ext_vector_type(8)));

#define DD 64
#define TILE_M 16

// Hardware fp32 scatter-add: force single global_atomic_add_f32 (no-return
// form, STOREcnt-tracked) instead of a possible CAS loop from the generic
// atomic builtin. CDNA5 L2 performs the RMW natively.
__device__ __forceinline__ void atomic_add_f32(float* p, float v) {
#if defined(__HIP_DEVICE_COMPILE__)
    asm volatile("global_atomic_add_f32 %0, %1, off"
                 :
                 : "v"(p), "v"(v)
                 : "memory");
#else
    (void)p; (void)v;
#endif
}

// ---------------------------------------------------------------------------
// K0: build ego = concat(user_emb, item_emb) and write layer-0 block of final
// (all_emb[0] is the raw, UN-normalized ego per the reference).
// one thread per float4 chunk of a node row (16 chunks per node)
// ---------------------------------------------------------------------------
__global__ void ngcf_init_kernel(const float* __restrict__ user_emb,
                                 const float* __restrict__ item_emb,
                                 float* __restrict__ ego,
                                 float* __restrict__ final_out,
                                 int n_users, int n_nodes) {
    int t = blockIdx.x * blockDim.x + threadIdx.x;
    int node = t >> 4;
    int q = t & 15;
    if (node >= n_nodes) return;
    float4 v;
    if (node < n_users)
        v = ((const float4*)user_emb)[(size_t)node * 16 + q];
    else
        v = ((const float4*)item_emb)[(size_t)(node - n_users) * 16 + q];
    ((float4*)ego)[(size_t)node * 16 + q] = v;
    // final is [N, 256] row-major -> 64 float4 per row; layer-0 block = chunks 0..15
    ((float4*)final_out)[(size_t)node * 64 + q] = v;
}

// ---------------------------------------------------------------------------
// K1: zero the side accumulator
// ---------------------------------------------------------------------------
__global__ void ngcf_zero_kernel(float4* __restrict__ p, int n4) {
    int t = blockIdx.x * blockDim.x + threadIdx.x;
    if (t < n4) p[t] = make_float4(0.f, 0.f, 0.f, 0.f);
}

// ---------------------------------------------------------------------------
// K2: edge-parallel SpMM scatter: side[row] += val * ego[col]
// 16 threads per edge, each handles one float4 chunk (4 hw f32 atomics).
// ego (38.4MB) is L2 resident on MI455X (192MB L2), so gathers mostly hit L2.
// ---------------------------------------------------------------------------
__global__ void ngcf_spmm_kernel(const float* __restrict__ ego,
                                 const float* __restrict__ vals,
                                 const int* __restrict__ rows,
                                 const int* __restrict__ cols,
                                 float* __restrict__ side, int nnz) {
    int t = blockIdx.x * blockDim.x + threadIdx.x;
    int e = t >> 4;
    int q = t & 15;
    if (e >= nnz) return;
    int r = rows[e];
    int c = cols[e];
    float v = vals[e];
    float4 g = ((const float4*)ego)[(size_t)c * 16 + q];
    float* dst = side + (size_t)r * DD + q * 4;
    atomic_add_f32(dst + 0, v * g.x);
    atomic_add_f32(dst + 1, v * g.y);
    atomic_add_f32(dst + 2, v * g.z);
    atomic_add_f32(dst + 3, v * g.w);
}

// ---------------------------------------------------------------------------
// K3: fused dense layer using V_WMMA_F32_16X16X4_F32 (exact fp32 GEMM).
//   sum = leakyrelu(side @ Wgc + bgc); bi = leakyrelu((ego*side) @ Wbi + bbi)
//   ego_new = sum + bi  -> written in-place to ego and (unnormalized) to final
// Block = 128 threads = 4 waves; wave w owns output columns [16w, 16w+16).
// Both 64x64 W matrices staged in LDS once, then B fragments live in VGPRs.
// Grid-stride loop over 16-row M-tiles.
// ---------------------------------------------------------------------------
__global__ __launch_bounds__(128)
void ngcf_dense_wmma_kernel(const float* __restrict__ ego_in,
                            const float* __restrict__ side,
                            const float* __restrict__ Wgc,
                            const float* __restrict__ bgc,
                            const float* __restrict__ Wbi,
                            const float* __restrict__ bbi,
                            float* __restrict__ ego_out,
                            float* __restrict__ final_out,
                            int n_nodes, int final_col_off) {
    __shared__ float sW[2 * DD * DD];  // 32 KB: [0..4095]=Wgc, [4096..8191]=Wbi

    const int tid = threadIdx.x;
    // cooperative LDS fill (float4): 2048 chunks / 128 threads = 16 each
    for (int idx = tid; idx < 2 * DD * DD / 4; idx += 128) {
        float4 a;
        if (idx < DD * DD / 4) a = ((const float4*)Wgc)[idx];
        else                   a = ((const float4*)Wbi)[idx - DD * DD / 4];
        ((float4*)sW)[idx] = a;
    }
    __syncthreads();

    const int wave  = tid >> 5;       // 0..3 : 16-column N-tile
    const int lane  = tid & 31;
    const int l15   = lane & 15;
    const int khalf = lane >> 4;      // 0 or 1: which K-pair of the 4-wide step
    const int ncol  = wave * 16 + l15;  // output column 0..63

    // B fragments for all 16 K-steps, both GEMMs, held in VGPRs.
    // Layout: lane holds B[kb][ncol], B[kb+1][ncol] with kb = 4*kk + 2*khalf.
    float bg0[16], bg1[16], bb0[16], bb1[16];
#pragma unroll
    for (int kk = 0; kk < 16; ++kk) {
        int kb = 4 * kk + 2 * khalf;
        bg0[kk] = sW[kb * DD + ncol];
        bg1[kk] = sW[(kb + 1) * DD + ncol];
        bb0[kk] = sW[DD * DD + kb * DD + ncol];
        bb1[kk] = sW[DD * DD + (kb + 1) * DD + ncol];
    }

    const float bias_g = bgc[ncol];
    const float bias_b = bbi[ncol];

    const int num_tiles = (n_nodes + TILE_M - 1) / TILE_M;
    for (int tile = blockIdx.x; tile < num_tiles; tile += gridDim.x) {
        int row = tile * TILE_M + l15;          // A-fragment row for this lane
        if (row >= n_nodes) row = n_nodes - 1;  // clamp (N%16==0 normally)
        const float* sr = side   + (size_t)row * DD;
        const float* er = ego_in + (size_t)row * DD;

        // prefetch next tile's A rows into cache (global_prefetch_b8)
        {
            int nt = tile + gridDim.x;
            if (nt < num_tiles) {
                int prow = nt * TILE_M + l15;
                if (prow < n_nodes) {
                    __builtin_prefetch(side   + (size_t)prow * DD, 0, 3);
                    __builtin_prefetch(ego_in + (size_t)prow * DD, 0, 3);
                }
            }
        }

        v8f acc_g = {0.f, 0.f, 0.f, 0.f, 0.f, 0.f, 0.f, 0.f};
        v8f acc_b = {0.f, 0.f, 0.f, 0.f, 0.f, 0.f, 0.f, 0.f};

#pragma unroll
        for (int kk = 0; kk < 16; ++kk) {
            int kb = 4 * kk + 2 * khalf;
            v2f s2 = *(const v2f*)(sr + kb);   // A_gc fragment: side[m][kb..kb+1]
            v2f e2 = *(const v2f*)(er + kb);
            v2f a_b = s2 * e2;                 // A_bi fragment: (ego*side)
            v2f b_g = {bg0[kk], bg1[kk]};
            v2f b_b = {bb0[kk], bb1[kk]};
            acc_g = __builtin_amdgcn_wmma_f32_16x16x4_f32(
                false, s2, false, b_g, (short)0, acc_g, false, false);
            acc_b = __builtin_amdgcn_wmma_f32_16x16x4_f32(
                false, a_b, false, b_b, (short)0, acc_b, false, false);
        }

        // epilogue: bias + leaky-relu(0.2) + sum, scatter stores
        // C/D layout: v[r] -> M = r + 8*khalf, N = ncol
#pragma unroll
        for (int r = 0; r < 8; ++r) {
            float xg = acc_g[r] + bias_g;
            float xb = acc_b[r] + bias_b;
            xg = (xg >= 0.f) ? xg : 0.2f * xg;
            xb = (xb >= 0.f) ? xb : 0.2f * xb;
            float y = xg + xb;
            int m = r + 8 * khalf;
            size_t node = (size_t)tile * TILE_M + m;
            if (node < (size_t)n_nodes) {
                ego_out[node * DD + ncol] = y;                      // un-normalized
                final_out[node * 256 + final_col_off + ncol] = y;  // normalized later
            }
        }
    }
}

// ---------------------------------------------------------------------------
// K4: L2-normalize one 64-wide column block of final, per row (eps=1e-12).
// one thread per node row: 16 float4 loads -> norm -> 16 float4 stores.
// ---------------------------------------------------------------------------
__global__ void ngcf_rownorm_kernel(float* __restrict__ final_out,
                                    int n_nodes, int final_col_off) {
    int n = blockIdx.x * blockDim.x + threadIdx.x;
    if (n >= n_nodes) return;
    float4* p = (float4*)(final_out + (size_t)n * 256 + final_col_off);
    float4 buf[16];
    float ss = 0.f;
#pragma unroll
    for (int c = 0; c < 16; ++c) {
        buf[c] = p[c];
        ss += buf[c].x * buf[c].x + buf[c].y * buf[c].y +
              buf[c].z * buf[c].z + buf[c].w * buf[c].w;
    }
    float inv = 1.f / fmaxf(sqrtf(ss), 1e-12f);
#pragma unroll
    for (int c = 0; c < 16; ++c) {
        buf[c].x *= inv; buf[c].y *= inv; buf[c].z *= inv; buf[c].w *= inv;
        p[c] = buf[c];
    }
}

// ---------------------------------------------------------------------------
// K5: gather u_g / i_g / j_g rows of final into d_out (concatenated flat).
// one thread per float4 chunk of an output row (64 chunks/row).
// ---------------------------------------------------------------------------
__global__ void ngcf_gather_kernel(const float* __restrict__ final_in,
                                   const int* __restrict__ u,
                                   const int* __restrict__ i,
                                   const int* __restrict__ j,
                                   float* __restrict__ out,
                                   int Bn, int n_users) {
    int t = blockIdx.x * blockDim.x + threadIdx.x;
    int r = t >> 6;
    int q = t & 63;
    if (r >= 3 * Bn) return;
    int src;
    if (r < Bn)          src = u[r];
    else if (r < 2 * Bn) src = n_users + i[r - Bn];
    else                 src = n_users + j[r - 2 * Bn];
    ((float4*)out)[(size_t)r * 64 + q] =
        ((const float4*)final_in)[(size_t)src * 64 + q];
}

// ---------------------------------------------------------------------------
extern "C" void kernel_launch(void* const* d_in, const int* in_sizes, int n_in,
                              void* d_out, int out_size, void* d_ws, size_t ws_size,
                              hipStream_t stream) {
    const float* user_emb = (const float*)d_in[0];
    const float* item_emb = (const float*)d_in[1];
    const float* W_gc     = (const float*)d_in[2];
    const float* b_gc     = (const float*)d_in[3];
    const float* W_bi     = (const float*)d_in[4];
    const float* b_bi     = (const float*)d_in[5];
    const float* adj_vals = (const float*)d_in[6];
    const int*   adj_rows = (const int*)d_in[7];
    const int*   adj_cols = (const int*)d_in[8];
    const int*   u_idx    = (const int*)d_in[9];
    const int*   i_idx    = (const int*)d_in[10];
    const int*   j_idx    = (const int*)d_in[11];

    const int n_users = in_sizes[0] / DD;
    const int n_items = in_sizes[1] / DD;
    const int n_nodes = n_users + n_items;
    const int n_layers = in_sizes[2] / (DD * DD);
    const int nnz = in_sizes[6];
    const int Bn = in_sizes[9];

    float* ego     = (float*)d_ws;                       // [N, 64]
    float* side    = ego + (size_t)n_nodes * DD;         // [N, 64]
    float* finalb  = side + (size_t)n_nodes * DD;        // [N, 256]
    float* out     = (float*)d_out;

    // K0: init ego + final layer-0 block
    {
        int total = n_nodes * 16;
        ngcf_init_kernel<<<(total + 255) / 256, 256, 0, stream>>>(
            user_emb, item_emb, ego, finalb, n_users, n_nodes);
    }

    for (int l = 0; l < n_layers; ++l) {
        // K1: zero side
        {
            int n4 = n_nodes * 16;
            ngcf_zero_kernel<<<(n4 + 255) / 256, 256, 0, stream>>>(
                (float4*)side, n4);
        }
        // K2: SpMM scatter (hardware global_atomic_add_f32)
        {
            long long total = (long long)nnz * 16;
            int blocks = (int)((total + 255) / 256);
            ngcf_spmm_kernel<<<blocks, 256, 0, stream>>>(
                ego, adj_vals, adj_rows, adj_cols, side, nnz);
        }
        // K3: fused dense WMMA layer (in-place ego update + final block write)
        {
            ngcf_dense_wmma_kernel<<<1024, 128, 0, stream>>>(
                ego, side,
                W_gc + (size_t)l * DD * DD, b_gc + (size_t)l * DD,
                W_bi + (size_t)l * DD * DD, b_bi + (size_t)l * DD,
                ego, finalb, n_nodes, DD * (l + 1));
        }
        // K4: L2 row-normalize the just-written final block
        {
            ngcf_rownorm_kernel<<<(n_nodes + 255) / 256, 256, 0, stream>>>(
                finalb, n_nodes, DD * (l + 1));
        }
    }

    // K5: gather outputs
    {
        int total = 3 * Bn * 64;
        ngcf_gather_kernel<<<(total + 255) / 256, 256, 0, stream>>>(
            finalb, u_idx, i_idx, j_idx, out, Bn, n_users);
    }
}